// HybridGCNGraphSAGE_2516850835929
// MI455X (gfx1250) — compile-verified
//
#include <hip/hip_runtime.h>

typedef __attribute__((ext_vector_type(2))) float v2f;
typedef __attribute__((ext_vector_type(8))) float v8f;

#define FDIM 64

// ---------------------------------------------------------------------------
// Degree / count
// ---------------------------------------------------------------------------
__global__ void init_cnt_kernel(float* __restrict__ cnt, int n) {
    int i = blockIdx.x * blockDim.x + threadIdx.x;
    if (i < n) cnt[i] = 0.0f;
}

__global__ void count_edges_kernel(const int* __restrict__ dst, float* __restrict__ cnt, int e) {
    int i = blockIdx.x * blockDim.x + threadIdx.x;
    if (i < e) atomicAdd(&cnt[dst[i]], 1.0f);
}

// dinv = rsqrt(cnt+1) (GCN: +1 self loop); sinv = 1/max(cnt,1) (SAGE mean)
__global__ void finalize_deg_kernel(const float* __restrict__ cnt, float* __restrict__ dinv,
                                    float* __restrict__ sinv, int n) {
    int i = blockIdx.x * blockDim.x + threadIdx.x;
    if (i < n) {
        float c = cnt[i];
        dinv[i] = rsqrtf(c + 1.0f);
        sinv[i] = 1.0f / fmaxf(c, 1.0f);
    }
}

// per-edge GCN norm weight
__global__ void edge_w_kernel(const int* __restrict__ src, const int* __restrict__ dst,
                              const float* __restrict__ dinv, float* __restrict__ ew, int e) {
    int i = blockIdx.x * blockDim.x + threadIdx.x;
    if (i < e) ew[i] = dinv[src[i]] * dinv[dst[i]];
}

// ---------------------------------------------------------------------------
// Fused WMMA GEMM:
//   out  = act( LN?(relu?(rowscale .* A)) @ W1  [+ LN?(B) @ W2]  + bias )
//   out2 = acc * dinv2[row]^2 + bias2                (optional, GCN self-loop init)
// A,B: (n x 64) row-major.  W1,W2: (64 x 64) row-major.
// One wave32 = one 16x64 output tile via v_wmma_f32_16x16x4_f32.
// Block = 256 threads = 8 waves = 128 rows; weights + LN params staged in LDS.
// ---------------------------------------------------------------------------
__launch_bounds__(256)
__global__ void gemm64_wmma_kernel(const float* __restrict__ A, const float* __restrict__ W1g,
                                   const float* __restrict__ B, const float* __restrict__ W2g,
                                   const float* __restrict__ bias,
                                   const float* __restrict__ rowscale,
                                   const float* __restrict__ lnAg, const float* __restrict__ lnAb,
                                   const float* __restrict__ lnBg, const float* __restrict__ lnBb,
                                   const float* __restrict__ dinv2, const float* __restrict__ bias2,
                                   float* __restrict__ out2,
                                   float* __restrict__ out, int n, int reluOut, int reluA) {
    __shared__ float W1[FDIM * FDIM];
    __shared__ float W2[FDIM * FDIM];
    __shared__ float LNP[4][FDIM];   // [0]=gammaA [1]=betaA [2]=gammaB [3]=betaB

    const int t = threadIdx.x;
    {   // cooperative staging
        const float4* s1 = (const float4*)W1g;
        float4* l1 = (float4*)W1;
        for (int i = t; i < 1024; i += 256) l1[i] = s1[i];
        if (B) {
            const float4* s2 = (const float4*)W2g;
            float4* l2 = (float4*)W2;
            for (int i = t; i < 1024; i += 256) l2[i] = s2[i];
        }
        if (lnAg && t < FDIM) { LNP[0][t] = lnAg[t]; LNP[1][t] = lnAb[t]; }
        if (lnBg && t < FDIM) { LNP[2][t] = lnBg[t]; LNP[3][t] = lnBb[t]; }
    }
    __syncthreads();

    const int wave  = t >> 5;
    const int lane  = t & 31;
    const int m0    = (blockIdx.x * 8 + wave) * 16;
    if (m0 >= n) return;                 // wave-uniform exit (after barrier)
    const int lhalf = lane >> 4;         // 0: K%4 in {0,1}, 1: K%4 in {2,3}
    const int l15   = lane & 15;
    const int kOff  = lhalf * 2;

    int row = m0 + l15;                  // this lane's A/B fragment row
    if (row >= n) row = n - 1;           // clamp: loads stay valid, stores guarded

    const float rs = rowscale ? rowscale[row] : 1.0f;

    // ---- A fragments (16x4 f32 layout), with fused relu/rowscale/LN ----
    v2f afrag[16];
    {
        const float* ar = A + (size_t)row * FDIM + kOff;
        #pragma unroll
        for (int kb = 0; kb < 16; ++kb) {
            float a0 = ar[kb * 4 + 0];
            float a1 = ar[kb * 4 + 1];
            if (reluA) { a0 = fmaxf(a0, 0.0f); a1 = fmaxf(a1, 0.0f); }
            afrag[kb].x = a0 * rs;
            afrag[kb].y = a1 * rs;
        }
        if (lnAg) {   // row is held entirely by lane pair (l15, l15+16)
            float s = 0.0f, q = 0.0f;
            #pragma unroll
            for (int kb = 0; kb < 16; ++kb) {
                s += afrag[kb].x + afrag[kb].y;
                q += afrag[kb].x * afrag[kb].x + afrag[kb].y * afrag[kb].y;
            }
            s += __shfl_xor(s, 16, 32);
            q += __shfl_xor(q, 16, 32);
            const float mu  = s * (1.0f / 64.0f);
            const float inv = rsqrtf(fmaxf(q * (1.0f / 64.0f) - mu * mu, 0.0f) + 1e-5f);
            #pragma unroll
            for (int kb = 0; kb < 16; ++kb) {
                const int k0 = kb * 4 + kOff;
                afrag[kb].x = (afrag[kb].x - mu) * inv * LNP[0][k0]     + LNP[1][k0];
                afrag[kb].y = (afrag[kb].y - mu) * inv * LNP[0][k0 + 1] + LNP[1][k0 + 1];
            }
        }
    }
    // ---- B fragments ----
    v2f bfrag[16];
    if (B) {
        const float* br = B + (size_t)row * FDIM + kOff;
        #pragma unroll
        for (int kb = 0; kb < 16; ++kb) {
            bfrag[kb].x = br[kb * 4 + 0];
            bfrag[kb].y = br[kb * 4 + 1];
        }
        if (lnBg) {
            float s = 0.0f, q = 0.0f;
            #pragma unroll
            for (int kb = 0; kb < 16; ++kb) {
                s += bfrag[kb].x + bfrag[kb].y;
                q += bfrag[kb].x * bfrag[kb].x + bfrag[kb].y * bfrag[kb].y;
            }
            s += __shfl_xor(s, 16, 32);
            q += __shfl_xor(q, 16, 32);
            const float mu  = s * (1.0f / 64.0f);
            const float inv = rsqrtf(fmaxf(q * (1.0f / 64.0f) - mu * mu, 0.0f) + 1e-5f);
            #pragma unroll
            for (int kb = 0; kb < 16; ++kb) {
                const int k0 = kb * 4 + kOff;
                bfrag[kb].x = (bfrag[kb].x - mu) * inv * LNP[2][k0]     + LNP[3][k0];
                bfrag[kb].y = (bfrag[kb].y - mu) * inv * LNP[2][k0 + 1] + LNP[3][k0 + 1];
            }
        }
    }

    #pragma unroll
    for (int nb = 0; nb < 4; ++nb) {
        const int n0 = nb * 16;
        v8f acc = {0.f, 0.f, 0.f, 0.f, 0.f, 0.f, 0.f, 0.f};

        // preload all W1 fragments for this N-tile, then run the WMMA chain
        v2f wf[16];
        #pragma unroll
        for (int kb = 0; kb < 16; ++kb) {
            const int kr = kb * 4 + kOff;
            wf[kb].x = W1[kr * FDIM + n0 + l15];
            wf[kb].y = W1[(kr + 1) * FDIM + n0 + l15];
        }
        #pragma unroll
        for (int kb = 0; kb < 16; ++kb)
            acc = __builtin_amdgcn_wmma_f32_16x16x4_f32(
                false, afrag[kb], false, wf[kb], (short)0, acc, false, false);

        if (B) {
            v2f wg[16];
            #pragma unroll
            for (int kb = 0; kb < 16; ++kb) {
                const int kr = kb * 4 + kOff;
                wg[kb].x = W2[kr * FDIM + n0 + l15];
                wg[kb].y = W2[(kr + 1) * FDIM + n0 + l15];
            }
            #pragma unroll
            for (int kb = 0; kb < 16; ++kb)
                acc = __builtin_amdgcn_wmma_f32_16x16x4_f32(
                    false, bfrag[kb], false, wg[kb], (short)0, acc, false, false);
        }

        const float bval  = bias  ? bias[n0 + l15]  : 0.0f;
        const float b2val = bias2 ? bias2[n0 + l15] : 0.0f;
        // C/D layout: VGPR j -> M = j (lanes 0-15) / j+8 (lanes 16-31), N = n0+l15
        #pragma unroll
        for (int j = 0; j < 8; ++j) {
            const int r = m0 + j + lhalf * 8;
            if (r < n) {
                float v = acc[j] + bval;
                if (reluOut) v = fmaxf(v, 0.0f);
                out[(size_t)r * FDIM + n0 + l15] = v;
                if (out2) {
                    const float d = dinv2[r];
                    out2[(size_t)r * FDIM + n0 + l15] = acc[j] * d * d + b2val;
                }
            }
        }
    }
}

// ---------------------------------------------------------------------------
// Edge scatters: 16 threads per edge, float4 gather, 4 contiguous f32 atomics
// ---------------------------------------------------------------------------
__global__ void scatter_gcn_kernel(const float* __restrict__ h, const int* __restrict__ src,
                                   const int* __restrict__ dst, const float* __restrict__ ew,
                                   float* __restrict__ g, int e) {
    long long tid = (long long)blockIdx.x * blockDim.x + threadIdx.x;
    long long edge = tid >> 4;
    if (edge >= e) return;
    int q = (int)(tid & 15);
    int s = src[edge], d = dst[edge];
    float w = ew[edge];
    float4 hv = ((const float4*)(h + (size_t)s * FDIM))[q];
    float* gp = g + (size_t)d * FDIM + q * 4;
    atomicAdd(gp + 0, hv.x * w);
    atomicAdd(gp + 1, hv.y * w);
    atomicAdd(gp + 2, hv.z * w);
    atomicAdd(gp + 3, hv.w * w);
}

__global__ void scatter_sum_kernel(const float* __restrict__ xin, const int* __restrict__ src,
                                   const int* __restrict__ dst, float* __restrict__ agg, int e) {
    long long tid = (long long)blockIdx.x * blockDim.x + threadIdx.x;
    long long edge = tid >> 4;
    if (edge >= e) return;
    int q = (int)(tid & 15);
    int s = src[edge], d = dst[edge];
    float4 xv = ((const float4*)(xin + (size_t)s * FDIM))[q];
    float* ap = agg + (size_t)d * FDIM + q * 4;
    atomicAdd(ap + 0, xv.x);
    atomicAdd(ap + 1, xv.y);
    atomicAdd(ap + 2, xv.z);
    atomicAdd(ap + 3, xv.w);
}

__global__ void zero_kernel(float* __restrict__ p, long long total) {
    long long i = (long long)blockIdx.x * blockDim.x + threadIdx.x;
    if (i < total) p[i] = 0.0f;
}

// ---------------------------------------------------------------------------
extern "C" void kernel_launch(void* const* d_in, const int* in_sizes, int n_in,
                              void* d_out, int out_size, void* d_ws, size_t ws_size,
                              hipStream_t stream) {
    const float* x        = (const float*)d_in[0];
    const int*   ei       = (const int*)d_in[1];
    const float* gcn_w1   = (const float*)d_in[2];
    const float* gcn_b1   = (const float*)d_in[3];
    const float* gcn_w2   = (const float*)d_in[4];
    const float* gcn_b2   = (const float*)d_in[5];
    const float* sage_wl1 = (const float*)d_in[6];
    const float* sage_bl1 = (const float*)d_in[7];
    const float* sage_wr1 = (const float*)d_in[8];
    const float* sage_wl2 = (const float*)d_in[9];
    const float* sage_bl2 = (const float*)d_in[10];
    const float* sage_wr2 = (const float*)d_in[11];
    const float* gcn_ln_g = (const float*)d_in[12];
    const float* gcn_ln_b = (const float*)d_in[13];
    const float* sage_ln_g = (const float*)d_in[14];
    const float* sage_ln_b = (const float*)d_in[15];
    const float* proj_w   = (const float*)d_in[16];
    const float* proj_b   = (const float*)d_in[17];

    const int N = in_sizes[0] / FDIM;
    const int E = in_sizes[1] / 2;
    const int* src = ei;
    const int* dst = ei + E;

    // workspace layout
    float* cnt  = (float*)d_ws;                 // N   (degree count)
    float* dinv = cnt + N;                      // N
    float* sinv = dinv + N;                     // N
    float* ew   = sinv + N;                     // E   (GCN edge weights)
    float* h    = ew + E;                       // N*64  scratch / final SAGE out
    float* g    = h + (size_t)N * FDIM;         // N*64  GCN branch
    float* agg  = g + (size_t)N * FDIM;         // N*64  SAGE aggregation
    float* out  = (float*)d_out;                // N*64  final out; doubles as SAGE s1

    const long long NM = (long long)N * FDIM;
    const int nblkN  = (N + 255) / 256;
    const int nblkE  = (E + 255) / 256;
    const int nblkNM = (int)((NM + 255) / 256);
    const int nblkSc = (int)(((long long)E * 16 + 255) / 256);
    const int nblkG  = (N + 127) / 128;         // gemm: 128 rows/block

    // degrees + edge weights
    init_cnt_kernel<<<nblkN, 256, 0, stream>>>(cnt, N);
    count_edges_kernel<<<nblkE, 256, 0, stream>>>(dst, cnt, E);
    finalize_deg_kernel<<<nblkN, 256, 0, stream>>>(cnt, dinv, sinv, N);
    edge_w_kernel<<<nblkE, 256, 0, stream>>>(src, dst, dinv, ew, E);

    // ---- GCN layer 1: h = x@W1 ; g = h*dinv^2 + b1 (fused) ; scatter ----
    gemm64_wmma_kernel<<<nblkG, 256, 0, stream>>>(
        x, gcn_w1, nullptr, nullptr, nullptr, nullptr,
        nullptr, nullptr, nullptr, nullptr,
        dinv, gcn_b1, g, h, N, 0, 0);
    scatter_gcn_kernel<<<nblkSc, 256, 0, stream>>>(h, src, dst, ew, g, E);

    // ---- GCN layer 2: relu fused into A-load; in-place out2=g ----
    gemm64_wmma_kernel<<<nblkG, 256, 0, stream>>>(
        g, gcn_w2, nullptr, nullptr, nullptr, nullptr,
        nullptr, nullptr, nullptr, nullptr,
        dinv, gcn_b2, g, h, N, 0, 1 /*reluA*/);
    scatter_gcn_kernel<<<nblkSc, 256, 0, stream>>>(h, src, dst, ew, g, E);

    // ---- SAGE layer 1: s1(=d_out) = relu( mean_agg(x)@Wl1 + x@Wr1 + bl1 ) ----
    zero_kernel<<<nblkNM, 256, 0, stream>>>(agg, NM);
    scatter_sum_kernel<<<nblkSc, 256, 0, stream>>>(x, src, dst, agg, E);
    gemm64_wmma_kernel<<<nblkG, 256, 0, stream>>>(
        agg, sage_wl1, x, sage_wr1, sage_bl1, sinv,
        nullptr, nullptr, nullptr, nullptr,
        nullptr, nullptr, nullptr, out, N, 1 /*reluOut*/, 0);

    // ---- SAGE layer 2: s2(=h) = mean_agg(s1)@Wl2 + s1@Wr2 + bl2 ----
    zero_kernel<<<nblkNM, 256, 0, stream>>>(agg, NM);
    scatter_sum_kernel<<<nblkSc, 256, 0, stream>>>(out, src, dst, agg, E);
    gemm64_wmma_kernel<<<nblkG, 256, 0, stream>>>(
        agg, sage_wl2, out, sage_wr2, sage_bl2, sinv,
        nullptr, nullptr, nullptr, nullptr,
        nullptr, nullptr, nullptr, h, N, 0, 0);

    // ---- fused LN(g), LN(s2) + concat-projection:
    //      out = LN(g)@Ptop + LN(s2)@Pbot + pb ----
    gemm64_wmma_kernel<<<nblkG, 256, 0, stream>>>(
        g, proj_w, h, proj_w + FDIM * FDIM, proj_b, nullptr,
        gcn_ln_g, gcn_ln_b, sage_ln_g, sage_ln_b,
        nullptr, nullptr, nullptr, out, N, 0, 0);
}